// ResGraph_695784702371
// MI455X (gfx1250) — compile-verified
//
#include <hip/hip_runtime.h>
#include <hip/hip_bf16.h>

typedef float v2f __attribute__((ext_vector_type(2)));
typedef float v8f __attribute__((ext_vector_type(8)));

#define HDIM 128
#define LEAKY 0.2f
#define PSTRIDE 144   // pair stride: 144 mod 32 == 16 -> conflict-free half-wave interleave

// ---------------- Kernel 0: s[h] = sum_k wh[h][k]  (row sums of wh) --------
__global__ void rowsum_kernel(const float* __restrict__ wh, float* __restrict__ s) {
    const int r = threadIdx.x;
    if (r < HDIM) {
        float acc = 0.f;
        #pragma unroll 8
        for (int k = 0; k < HDIM; ++k) acc += wh[r * HDIM + k];
        s[r] = acc;
    }
}

// ---------------- Kernel 1: h = x @ wi via V_WMMA_F32_16X16X4_F32 ----------
// 128 threads = 4 waves per block; each wave computes 16 rows x 128 cols.
// A fragment (16x4 f32): lanes 0-15 hold row=lane,   K = 4kk+{0,1};
//                        lanes 16-31 hold row=lane-16, K = 4kk+{2,3}.
// B staged in LDS as float2 pairs: pair(r2, col) = (wi[2*r2][col], wi[2*r2+1][col]),
// pair index = r2*PSTRIDE + col  -> one aligned ds_load_b64 per fragment,
// bank-pair = (r2*144+col) mod 32 == (c0+lane) mod 32 for all 32 lanes (conflict-free).
// C/D (16x16 f32, 8 VGPRs): VGPR i -> M=i (lanes 0-15) / M=i+8 (lanes 16-31), N=lane%16.
__global__ void __launch_bounds__(128)
gemm_wmma_kernel(const float* __restrict__ x, const float* __restrict__ wi,
                 float* __restrict__ hout, long Mrows) {
    __shared__ float lws[64 * PSTRIDE * 2];   // 72 KB
    const int tid = threadIdx.x;

    // cooperative pair-swizzled LDS fill of wi: 8192 pairs, 64 per thread
    #pragma unroll
    for (int it = 0; it < 64; ++it) {
        const int pid = it * 128 + tid;
        const int r2  = pid >> 7;        // wi row pair (rows 2*r2, 2*r2+1)
        const int col = pid & 127;
        lws[(r2 * PSTRIDE + col) * 2 + 0] = wi[(2 * r2 + 0) * HDIM + col];
        lws[(r2 * PSTRIDE + col) * 2 + 1] = wi[(2 * r2 + 1) * HDIM + col];
    }
    __syncthreads();

    const int wave  = tid >> 5;
    const int lane  = tid & 31;
    const int rsel  = lane & 15;
    const int khalf = lane >> 4;   // 0: K%4 in {0,1}; 1: K%4 in {2,3}

    const long rowStart = ((long)blockIdx.x * 4 + wave) * 16;
    if (rowStart >= Mrows) return;   // wave-uniform; after barrier

    // Load full-K A fragments for this wave's 16 rows (64 VGPRs)
    const float* arow = x + (rowStart + rsel) * (long)HDIM;
    v2f afrag[32];
    #pragma unroll
    for (int kk = 0; kk < 32; ++kk) {
        const float2 t = *(const float2*)(arow + kk * 4 + khalf * 2);
        afrag[kk].x = t.x;
        afrag[kk].y = t.y;
    }

    #pragma unroll
    for (int nt = 0; nt < 8; ++nt) {
        v8f acc = {};
        const int col = nt * 16 + rsel;
        #pragma unroll
        for (int kk = 0; kk < 32; ++kk) {
            // pair row r2 = 2*kk + khalf holds (K=4kk+2*khalf, K=4kk+2*khalf+1)
            const v2f bfrag = *(const v2f*)&lws[((2 * kk + khalf) * PSTRIDE + col) * 2];
            acc = __builtin_amdgcn_wmma_f32_16x16x4_f32(
                false, afrag[kk], false, bfrag, (short)0, acc, false, false);
        }
        #pragma unroll
        for (int i = 0; i < 8; ++i) {
            const long row = rowStart + i + khalf * 8;
            hout[row * HDIM + col] = acc[i];
        }
    }
}

// ---------------- Kernel 2: gather, diff, bitonic sort(128), dot(s), tanh --
// One wave per (b,e) row. Element index = q*32 + lane (q = 0..3).
__global__ void __launch_bounds__(256)
edge_sort_kernel(const float* __restrict__ h, const long long* __restrict__ ei,
                 const float* __restrict__ s, float* __restrict__ out,
                 int Nn, int Rows) {
    const int lane = threadIdx.x & 31;
    const int wid  = (int)((blockIdx.x * blockDim.x + threadIdx.x) >> 5);
    if (wid >= Rows) return;

    const int b = wid / Nn;
    const int e = wid - b * Nn;
    const long long nj = ei[e];        // edge_index[0][e] (source)
    const long long ni = ei[Nn + e];   // edge_index[1][e] (target)

    const float* hj = h + ((long)b * Nn + nj) * HDIM;
    const float* hi = h + ((long)b * Nn + ni) * HDIM;
    const float* hs = h + ((long)b * Nn + e)  * HDIM;

    float v[4];
    #pragma unroll
    for (int q = 0; q < 4; ++q) {
        const int idx = q * 32 + lane;
        v[q] = hj[idx] - hi[idx];
    }

    // Bitonic sort of 128 values striped as (q*32 + lane).
    #pragma unroll
    for (int k = 2; k <= 128; k <<= 1) {
        #pragma unroll
        for (int d = k >> 1; d > 0; d >>= 1) {
            if (d >= 32) {                       // partner in-lane (register pair)
                const int qd = d >> 5;
                #pragma unroll
                for (int q = 0; q < 4; ++q) {
                    if ((q & qd) == 0) {
                        const int  idx = q * 32 + lane;
                        const bool asc = (idx & k) == 0;
                        const float a  = v[q], bb = v[q | qd];
                        const float lo = fminf(a, bb), hi2 = fmaxf(a, bb);
                        v[q]      = asc ? lo  : hi2;
                        v[q | qd] = asc ? hi2 : lo;
                    }
                }
            } else {                             // partner cross-lane (ds permute)
                #pragma unroll
                for (int q = 0; q < 4; ++q) {
                    const int   idx   = q * 32 + lane;
                    const float other = __shfl_xor(v[q], d, 32);
                    const bool  lower = (lane & d) == 0;
                    const bool  asc   = (idx & k) == 0;
                    v[q] = (lower == asc) ? fminf(v[q], other) : fmaxf(v[q], other);
                }
            }
        }
    }

    // sum_k(d @ wh) == sorted_diff . rowsum(wh)
    float acc = 0.f;
    #pragma unroll
    for (int q = 0; q < 4; ++q) acc += v[q] * s[q * 32 + lane];
    #pragma unroll
    for (int d2 = 16; d2 >= 1; d2 >>= 1) acc += __shfl_xor(acc, d2, 32);

    float* orow = out + (long)wid * HDIM;
    #pragma unroll
    for (int q = 0; q < 4; ++q) {
        const int   idx = q * 32 + lane;
        const float hv  = hs[idx];
        orow[idx] = (1.0f - LEAKY) * hv + LEAKY * tanhf(acc + hv);
    }
}

extern "C" void kernel_launch(void* const* d_in, const int* in_sizes, int n_in,
                              void* d_out, int out_size, void* d_ws, size_t ws_size,
                              hipStream_t stream) {
    const float*     x  = (const float*)d_in[0];
    const long long* ei = (const long long*)d_in[1];   // int64 edge_index [2,N]
    const float*     wi = (const float*)d_in[2];
    const float*     wh = (const float*)d_in[3];
    float*           out = (float*)d_out;

    const int  Nn    = in_sizes[1] / 2;                      // 100000
    const long total = (long)in_sizes[0];                    // B*N*H
    const int  Bb    = (int)(total / ((long)Nn * HDIM));     // 4
    const long Mrows = (long)Bb * Nn;                        // 400000

    float* hbuf = (float*)d_ws;                   // [Mrows, 128] fp32 (~205 MB)
    float* sbuf = hbuf + (size_t)Mrows * HDIM;    // [128] fp32

    rowsum_kernel<<<1, 128, 0, stream>>>(wh, sbuf);

    const int gblocks = (int)((Mrows + 63) / 64);            // 6250
    gemm_wmma_kernel<<<gblocks, 128, 0, stream>>>(x, wi, hbuf, Mrows);

    const int rows    = (int)Mrows;
    const int eblocks = (int)(((long)rows * 32 + 255) / 256);  // 50000
    edge_sort_kernel<<<eblocks, 256, 0, stream>>>(hbuf, ei, sbuf, out, Nn, rows);
}